// MultiLevelAlignedRoIPooling_17188459118945
// MI455X (gfx1250) — compile-verified
//
#include <hip/hip_runtime.h>
#include <math.h>

// RoIAlign (multi-level) for MI455X / gfx1250.
// One block (256 threads = 8 wave32) per box. Gather the 14x14 bilinear
// sample grid into LDS (channel chunks of 64) — using gfx1250 async
// global->LDS loads when available — then compute
// out[49,256] = W[49,196] @ G[196,256] with V_WMMA_F32_16X16X4_F32.
// W has exactly 4 nonzeros per row at two even-aligned K pairs, so the
// A operand is synthesized branchlessly with two integer compares.

typedef float v2f  __attribute__((ext_vector_type(2)));
typedef float v8f  __attribute__((ext_vector_type(8)));
typedef int   v2i_t __attribute__((ext_vector_type(2)));

#define AS1 __attribute__((address_space(1)))
#define AS3 __attribute__((address_space(3)))

#if __has_builtin(__builtin_amdgcn_global_load_async_to_lds_b64)
#define HAVE_ASYNC_LDS 1
#else
#define HAVE_ASYNC_LDS 0
#endif

__device__ __forceinline__ void wait_asynccnt0() {
#if __has_builtin(__builtin_amdgcn_s_wait_asynccnt)
    __builtin_amdgcn_s_wait_asynccnt(0);
#else
    asm volatile("s_wait_asynccnt 0" ::: "memory");
#endif
}

#define OUTD   7
#define SAMP   14        // 2*OUTD sample lines per axis
#define KDIM   196       // 14*14 gathered points = WMMA K dimension
#define KSTEPS 49        // 196 / 4
#define CCHUNK 64        // channels per LDS tile
#define GPAD   72        // padded row stride (2*72 mod 64 = 16 -> halves hit disjoint banks)

__global__ __launch_bounds__(256)
void roialign_wmma_kernel(const float* __restrict__ f0, const float* __restrict__ f1,
                          const float* __restrict__ f2, const float* __restrict__ f3,
                          const float* __restrict__ f4, const float* __restrict__ boxes,
                          float* __restrict__ out, int Nboxes, int C)
{
    __shared__ float G[KDIM][GPAD];      // gathered points x channel chunk
    __shared__ float kyv[SAMP], kxv[SAMP];
    __shared__ int   yix[SAMP], xix[SAMP];
    __shared__ int   poff[KDIM];         // (y*S + x) * C per sample point

    const int box = blockIdx.x;          // flat b*N + n
    const int tid = threadIdx.x;

    // ---- per-box parameters (uniform across block, computed redundantly) ----
    const float* bp = boxes + (size_t)box * 4;
    const float y1 = bp[0], x1 = bp[1], y2 = bp[2], x2 = bp[3];
    const float bh = y2 - y1, bw = x2 - x1;
    const float asq = sqrtf(bh * bw);
    int levels = (int)floorf(log2f(asq * (1.0f / 224.0f))) + 4;
    levels = min(max(levels, 4), 64);
    const float scale = exp2f((float)levels);
    const int   lvl   = levels - 4;
    const float bnd   = 128.0f / exp2f((float)lvl) - 1.0f;
    const int   lvlc  = min(lvl, 4);     // clamp gather level like JAX OOB-clip

    const float* feat; int S;
    switch (lvlc) {
        case 0:  feat = f0; S = 128; break;
        case 1:  feat = f1; S = 64;  break;
        case 2:  feat = f2; S = 32;  break;
        case 3:  feat = f3; S = 16;  break;
        default: feat = f4; S = 8;   break;
    }
    const int b = box / Nboxes;
    const float* fbase = feat + (size_t)b * S * S * C;

    // ---- phase 1a: sample coordinates + bilinear weights into LDS ----
    if (tid < OUTD) {                      // y axis, samples 0..6
        const int k = tid;
        const float g  = (y1 / scale - 0.5f) + ((k + 0.5f) / (float)OUTD) * (bh / scale);
        const float p0 = fmaxf(0.0f, floorf(g));
        const float l  = g - p0;
        kyv[2 * k]     = 1.0f - l;
        kyv[2 * k + 1] = l;
        yix[2 * k]     = max((int)fminf(p0,        bnd), 0);
        yix[2 * k + 1] = max((int)fminf(p0 + 1.0f, bnd), 0);
    } else if (tid >= 16 && tid < 16 + OUTD) {  // x axis, samples 0..6
        const int k = tid - 16;
        const float g  = (x1 / scale - 0.5f) + ((k + 0.5f) / (float)OUTD) * (bw / scale);
        const float p0 = fmaxf(0.0f, floorf(g));
        const float l  = g - p0;
        kxv[2 * k]     = 1.0f - l;
        kxv[2 * k + 1] = l;
        xix[2 * k]     = max((int)fminf(p0,        bnd), 0);
        xix[2 * k + 1] = max((int)fminf(p0 + 1.0f, bnd), 0);
    }
    __syncthreads();

    // ---- phase 1b: per-point global offset table ----
    if (tid < KDIM) {
        const int i = tid / SAMP;
        const int j = tid - i * SAMP;
        poff[tid] = (yix[i] * S + xix[j]) * C;
    }

    // ---- wave / tile mapping ----
    const int wave  = tid >> 5;
    const int lane  = tid & 31;
    const int half  = lane >> 4;          // 0: lanes 0-15, 1: lanes 16-31
    const int l16   = lane & 15;
    const int mtile = wave & 3;           // 4 M-tiles cover p = 0..63 (49 valid)
    const int ct0   = wave >> 2;          // this wave owns ctiles {ct0, ct0+2}

    const int m    = mtile * 16 + l16;    // output row this lane feeds into A
    const int oy   = m / OUTD, ox = m - (m / OUTD) * OUTD;
    const bool mval = (m < 49);
    const int oyc  = min(oy, OUTD - 1);   // keep LDS reads in range for pad rows

    // ---- hoisted A-operand constants: W[m,.] nonzeros live at even pairs
    //      kA = 28*oy + 2*ox (weights wy0*wx0, wy0*wx1) and kB = kA + 14.
    const float wy0 = kyv[2 * oyc],     wy1 = kyv[2 * oyc + 1];
    const float wx0 = kxv[2 * ox],      wx1 = kxv[2 * ox + 1];
    const float v00 = mval ? wy0 * wx0 : 0.0f;
    const float v01 = mval ? wy0 * wx1 : 0.0f;
    const float v10 = mval ? wy1 * wx0 : 0.0f;
    const float v11 = mval ? wy1 * wx1 : 0.0f;
    const int   kA  = 2 * SAMP * oyc + 2 * ox;
    const int   kB  = kA + SAMP;

    const int colB0 = ct0 * 16 + l16;
    const int colB1 = (ct0 + 2) * 16 + l16;

    __syncthreads();                      // poff ready

    const size_t outbase = (size_t)box * 49 * C;

    for (int cc = 0; cc < C; cc += CCHUNK) {
        // ---- gather: 64 channels x 196 points, b64 per lane (256B/warp) ----
        {
            const int col2 = (tid & 31) * 2;
            const int ch   = cc + col2;
#if HAVE_ASYNC_LDS
            for (int s = tid >> 5; s < KDIM; s += 8) {
                const float* src = fbase + poff[s] + ch;
                __builtin_amdgcn_global_load_async_to_lds_b64(
                    (AS1 v2i_t*)src, (AS3 v2i_t*)(&G[s][col2]),
                    /*imm offset*/0, /*cpol*/0);
            }
            wait_asynccnt0();             // LDS writes complete before barrier
#else
            for (int s = tid >> 5; s < KDIM; s += 8) {
                const float2 v = *(const float2*)(fbase + poff[s] + ch);
                *(float2*)(&G[s][col2]) = v;
            }
#endif
        }
        __syncthreads();

        // ---- WMMA: D = W @ G, K = 196 in 49 steps of 4, branchless A ----
        v8f acc0 = {};
        v8f acc1 = {};
        int k0 = 2 * half;
        #pragma unroll 7
        for (int ks = 0; ks < KSTEPS; ++ks, k0 += 4) {
            const bool eA = (k0 == kA);
            const bool eB = (k0 == kB);
            v2f a;
            a.x = eA ? v00 : (eB ? v10 : 0.0f);
            a.y = eA ? v01 : (eB ? v11 : 0.0f);

            v2f bv0, bv1;
            bv0.x = G[k0][colB0];     bv0.y = G[k0 + 1][colB0];
            bv1.x = G[k0][colB1];     bv1.y = G[k0 + 1][colB1];

            acc0 = __builtin_amdgcn_wmma_f32_16x16x4_f32(
                       false, a, false, bv0, (short)0, acc0, false, false);
            acc1 = __builtin_amdgcn_wmma_f32_16x16x4_f32(
                       false, a, false, bv1, (short)0, acc1, false, false);
        }

        // ---- store: D VGPR r -> M = r + 8*half, N = l16 ----
        #pragma unroll
        for (int r = 0; r < 8; ++r) {
            const int p = mtile * 16 + r + 8 * half;
            if (p < 49) {
                out[outbase + (size_t)p * C + cc + colB0] = acc0[r];
                out[outbase + (size_t)p * C + cc + colB1] = acc1[r];
            }
        }
        __syncthreads();   // before next chunk overwrites G
    }
}

extern "C" void kernel_launch(void* const* d_in, const int* in_sizes, int n_in,
                              void* d_out, int out_size, void* d_ws, size_t ws_size,
                              hipStream_t stream) {
    const float* f0    = (const float*)d_in[0];
    const float* f1    = (const float*)d_in[1];
    const float* f2    = (const float*)d_in[2];
    const float* f3    = (const float*)d_in[3];
    const float* f4    = (const float*)d_in[4];
    const float* boxes = (const float*)d_in[5];
    float* out = (float*)d_out;

    const int C = 256;
    const int total_boxes = in_sizes[5] / 4;             // B*N
    const int B = in_sizes[0] / (128 * 128 * C);
    const int N = total_boxes / B;

    roialign_wmma_kernel<<<total_boxes, 256, 0, stream>>>(
        f0, f1, f2, f3, f4, boxes, out, N, C);
}